// GPTNeoXAttention_55336358642526
// MI455X (gfx1250) — compile-verified
//
#include <hip/hip_runtime.h>
#include <math.h>

typedef __attribute__((ext_vector_type(2))) float v2f;
typedef __attribute__((ext_vector_type(4))) float f4;
typedef __attribute__((ext_vector_type(8))) float v8f;

#define BATCH   2
#define SEQ     2048
#define HIDDEN  2048
#define NHEADS  16
#define HEADSZ  128
#define QKVDIM  6144          // 3*HIDDEN, per-head layout: h*384 + {q:0..127, k:128..255, v:256..383}
#define ROTH    16            // rotary_ndims/2
#define MASKVAL (-3.4028234663852886e38f)

// ---------------------------------------------------------------------------
// Kernel 1/4: C[M,N] = A[M,K] * B[K,N] + bias, fp32 via V_WMMA_F32_16X16X4_F32
// Block tile 128x128, K-step 32, double-buffered LDS, B stored transposed
// [n][k] (stride 36) so every fragment is one aligned ds_load_b64.
// 8 waves, each wave: 2x4 16x16 accumulator tiles.
// ---------------------------------------------------------------------------
__global__ __launch_bounds__(256) void gemm_bias_wmma(
    const float* __restrict__ A, const float* __restrict__ B,
    const float* __restrict__ bias, float* __restrict__ C,
    int M, int N, int K)
{
    __shared__ float As[2][128 * 36];   // [row][k], stride 36: f4-store aligned, even for b64 frags
    __shared__ float Bt[2][128 * 36];   // [col][k] transposed, same stride

    const int tid  = threadIdx.x;
    const int lane = tid & 31;
    const int wid  = tid >> 5;
    const int lh   = lane & 15;          // M/N index within fragment
    const int kh2  = (lane >> 4) << 1;   // K sub-offset {0,2}
    const int rsel = (lane >> 4) << 3;   // row offset {0,8} for C/D layout

    const int m0 = blockIdx.y * 128;
    const int n0 = blockIdx.x * 128;
    const int waveRow = (wid & 3) * 32;
    const int waveCol = (wid >> 2) * 64;

    v8f zero = {};
    v8f acc[2][4];
    #pragma unroll
    for (int i = 0; i < 2; ++i)
        #pragma unroll
        for (int j = 0; j < 4; ++j) acc[i][j] = zero;

    // Per-thread staging decomposition (4 iterations each):
    //  A: e = it*256+tid -> row = e>>3, c4 = (e&7)<<2    (f4 global load, f4 LDS store)
    //  B: e = it*256+tid -> n = e&127, k4 = (e>>7)<<2    (4x coalesced b32 global, f4 LDS store)
    f4 ra[4], rb[4];

    const int nt = K >> 5;

    // prologue: tile 0 -> regs -> buffer 0
    #pragma unroll
    for (int it = 0; it < 4; ++it) {
        int e = it * 256 + tid;
        int r = e >> 3, c = (e & 7) << 2;
        ra[it] = *(const f4*)(A + (size_t)(m0 + r) * K + c);
        int n = e & 127, k4 = (e >> 7) << 2;
        f4 t;
        t.x = B[(size_t)(k4 + 0) * N + n0 + n];
        t.y = B[(size_t)(k4 + 1) * N + n0 + n];
        t.z = B[(size_t)(k4 + 2) * N + n0 + n];
        t.w = B[(size_t)(k4 + 3) * N + n0 + n];
        rb[it] = t;
    }
    #pragma unroll
    for (int it = 0; it < 4; ++it) {
        int e = it * 256 + tid;
        int r = e >> 3, c = (e & 7) << 2;
        *(f4*)(&As[0][r * 36 + c]) = ra[it];
        int n = e & 127, k4 = (e >> 7) << 2;
        *(f4*)(&Bt[0][n * 36 + k4]) = rb[it];
    }

    for (int t = 0; t < nt; ++t) {
        const int cur = t & 1;
        const int k0n = (t + 1) << 5;

        if (t + 1 < nt) {                 // prefetch next tile into registers
            #pragma unroll
            for (int it = 0; it < 4; ++it) {
                int e = it * 256 + tid;
                int r = e >> 3, c = (e & 7) << 2;
                ra[it] = *(const f4*)(A + (size_t)(m0 + r) * K + k0n + c);
                int n = e & 127, k4 = (e >> 7) << 2;
                f4 tv;
                tv.x = B[(size_t)(k0n + k4 + 0) * N + n0 + n];
                tv.y = B[(size_t)(k0n + k4 + 1) * N + n0 + n];
                tv.z = B[(size_t)(k0n + k4 + 2) * N + n0 + n];
                tv.w = B[(size_t)(k0n + k4 + 3) * N + n0 + n];
                rb[it] = tv;
            }
        }

        __syncthreads();                  // buffer `cur` fully staged

        const float* Ac = &As[cur][0];
        const float* Bc = &Bt[cur][0];
        #pragma unroll
        for (int kk = 0; kk < 32; kk += 4) {
            int kb = kk + kh2;
            v2f af[2];
            #pragma unroll
            for (int i = 0; i < 2; ++i)
                af[i] = *(const v2f*)(&Ac[(waveRow + i * 16 + lh) * 36 + kb]);
            v2f bf[4];
            #pragma unroll
            for (int j = 0; j < 4; ++j)
                bf[j] = *(const v2f*)(&Bc[(waveCol + j * 16 + lh) * 36 + kb]);
            #pragma unroll
            for (int i = 0; i < 2; ++i)
                #pragma unroll
                for (int j = 0; j < 4; ++j)
                    acc[i][j] = __builtin_amdgcn_wmma_f32_16x16x4_f32(
                        false, af[i], false, bf[j], (short)0, acc[i][j], false, false);
        }

        if (t + 1 < nt) {                 // stage next tile into the other buffer
            #pragma unroll
            for (int it = 0; it < 4; ++it) {
                int e = it * 256 + tid;
                int r = e >> 3, c = (e & 7) << 2;
                *(f4*)(&As[cur ^ 1][r * 36 + c]) = ra[it];
                int n = e & 127, k4 = (e >> 7) << 2;
                *(f4*)(&Bt[cur ^ 1][n * 36 + k4]) = rb[it];
            }
        }
    }

    // store D layout: VGPR r -> row (r + rsel), lane -> col lh
    #pragma unroll
    for (int i = 0; i < 2; ++i) {
        #pragma unroll
        for (int j = 0; j < 4; ++j) {
            int gcol = n0 + waveCol + j * 16 + lh;
            float bv = bias[gcol];
            #pragma unroll
            for (int r = 0; r < 8; ++r) {
                int grow = m0 + waveRow + i * 16 + r + rsel;
                C[(size_t)grow * N + gcol] = acc[i][j][r] + bv;
            }
        }
    }
}

// ---------------------------------------------------------------------------
// Kernel 2/4: in-place RoPE on q and k rotary dims inside the qkv buffer.
// One thread per (b, s, h, i) with i in [0,16): pairs (i, i+16).
// ---------------------------------------------------------------------------
__global__ __launch_bounds__(256) void rope_kernel(
    float* __restrict__ qkv, const int* __restrict__ pos_ids)
{
    int idx = blockIdx.x * 256 + threadIdx.x;     // B*S*H*16 = 2^20 threads
    int i = idx & 15;
    int h = (idx >> 4) & 15;
    int s = (idx >> 8) & (SEQ - 1);
    int b = idx >> 19;

    float t = (float)pos_ids[s];
    float freq = powf(10000.0f, -(float)(2 * i) / 32.0f);
    float sn, cs;
    sincosf(t * freq, &sn, &cs);

    size_t base = ((size_t)(b * SEQ + s)) * QKVDIM + (size_t)h * 384;
    float q0 = qkv[base + i], q1 = qkv[base + i + ROTH];
    qkv[base + i]        = q0 * cs - q1 * sn;
    qkv[base + i + ROTH] = q1 * cs + q0 * sn;
    float k0 = qkv[base + 128 + i], k1 = qkv[base + 128 + i + ROTH];
    qkv[base + 128 + i]        = k0 * cs - k1 * sn;
    qkv[base + 128 + i + ROTH] = k1 * cs + k0 * sn;
}

// ---------------------------------------------------------------------------
// Kernel 3/4: causal flash attention. 1 WG per (b, h, 64 q-rows), 4 waves,
// each wave owns 16 q-rows. fp32 WMMA for QK^T and P*V, online softmax.
// K staged row-major (b64 frags), V staged transposed [dim][key] (b64 frags),
// both double-buffered with register prefetch. ctx written [B,S,HIDDEN].
// ---------------------------------------------------------------------------
__global__ __launch_bounds__(128) void flash_attn_kernel(
    const float* __restrict__ qkv, const float* __restrict__ amask,
    float* __restrict__ ctx)
{
    __shared__ float Qs[4 * 16 * 132];     // 64 q-rows x 128 dims
    __shared__ float Ks[2][16 * 132];      // [key][dim]
    __shared__ float Vt[2][128 * 20];      // [dim][key], stride 20 (f4 aligned, even)
    __shared__ float Ps[4 * 16 * 18];      // per-wave 16x16 P tile (A-fragment layout)

    const int tid  = threadIdx.x;
    const int lane = tid & 31;
    const int w    = tid >> 5;
    const int lh   = lane & 15;
    const int kh2  = (lane >> 4) << 1;
    const int rsel = (lane >> 4) << 3;

    const int b  = blockIdx.y >> 4;
    const int h  = blockIdx.y & 15;
    const int q0 = blockIdx.x * 64;

    const size_t qkv_bh = (size_t)b * SEQ * QKVDIM + (size_t)h * 384;

    // stage Q tile 64x128 (row r of wave w at Qs[r*132])
    #pragma unroll
    for (int it = 0; it < 16; ++it) {
        int e = it * 128 + tid;
        int r = e >> 5;
        int c = (e & 31) << 2;
        f4 v = *(const f4*)(qkv + qkv_bh + (size_t)(q0 + r) * QKVDIM + c);
        *(f4*)(&Qs[r * 132 + c]) = v;
    }

    v8f zero = {};
    v8f o[8];
    #pragma unroll
    for (int j = 0; j < 8; ++j) o[j] = zero;
    float mrow[8], lrow[8];
    #pragma unroll
    for (int r = 0; r < 8; ++r) { mrow[r] = MASKVAL; lrow[r] = 0.0f; }

    const float norm = 0.08838834764831845f;   // 1/sqrt(128)
    const int kb_hi = (q0 + 63) >> 4;
    float* Qw = &Qs[w * 16 * 132];
    float* Pw = &Ps[w * 16 * 18];

    // K/V staging (4 iterations each, per thread):
    //  K: e = it*128+tid -> key = e>>5,  c4 = (e&31)<<2   (f4 load, f4 store row-major)
    //  V: e = it*128+tid -> dim = e&127, k4 = (e>>7)<<2   (4x coalesced b32, f4 store transposed)
    f4 rk[4], rv[4];

    auto loadKV = [&](int kblk) {
        #pragma unroll
        for (int it = 0; it < 4; ++it) {
            int e = it * 128 + tid;
            int kr = e >> 5, kc = (e & 31) << 2;
            rk[it] = *(const f4*)(qkv + qkv_bh + (size_t)(kblk * 16 + kr) * QKVDIM + 128 + kc);
            int vd = e & 127, vk = (e >> 7) << 2;
            f4 tv;
            tv.x = qkv[qkv_bh + (size_t)(kblk * 16 + vk + 0) * QKVDIM + 256 + vd];
            tv.y = qkv[qkv_bh + (size_t)(kblk * 16 + vk + 1) * QKVDIM + 256 + vd];
            tv.z = qkv[qkv_bh + (size_t)(kblk * 16 + vk + 2) * QKVDIM + 256 + vd];
            tv.w = qkv[qkv_bh + (size_t)(kblk * 16 + vk + 3) * QKVDIM + 256 + vd];
            rv[it] = tv;
        }
    };
    auto storeKV = [&](int buf) {
        #pragma unroll
        for (int it = 0; it < 4; ++it) {
            int e = it * 128 + tid;
            int kr = e >> 5, kc = (e & 31) << 2;
            *(f4*)(&Ks[buf][kr * 132 + kc]) = rk[it];
            int vd = e & 127, vk = (e >> 7) << 2;
            *(f4*)(&Vt[buf][vd * 20 + vk]) = rv[it];
        }
    };

    loadKV(0);
    storeKV(0);

    for (int kblk = 0; kblk <= kb_hi; ++kblk) {
        const int cur = kblk & 1;
        if (kblk < kb_hi) loadKV(kblk + 1);    // prefetch next block into registers

        __syncthreads();                       // buffer `cur` staged (and Qs on iter 0)

        // S = Q * K^T over d=128 (32 WMMA steps of K=4)
        v8f s = zero;
        const float* Kc = &Ks[cur][0];
        #pragma unroll
        for (int step = 0; step < 32; ++step) {
            int kd = step * 4 + kh2;
            v2f a   = *(const v2f*)(&Qw[lh * 132 + kd]);   // A: q-row=lh, dims kd,kd+1
            v2f bfr = *(const v2f*)(&Kc[lh * 132 + kd]);   // B=K^T: key=lh, dims kd,kd+1
            s = __builtin_amdgcn_wmma_f32_16x16x4_f32(false, a, false, bfr,
                                                      (short)0, s, false, false);
        }

        int   kidx = kblk * 16 + lh;
        float am   = amask[(size_t)b * SEQ + kidx];
        float sv[8];
        #pragma unroll
        for (int r = 0; r < 8; ++r) {
            int srow = q0 + w * 16 + r + rsel;
            float v = (kidx <= srow) ? (s[r] * norm) : MASKVAL;
            sv[r] = v + am;
        }

        // online softmax: per-row reduce across the 16 lanes holding the row
        #pragma unroll
        for (int r = 0; r < 8; ++r) {
            float mx = sv[r];
            mx = fmaxf(mx, __shfl_xor(mx, 1, 32));
            mx = fmaxf(mx, __shfl_xor(mx, 2, 32));
            mx = fmaxf(mx, __shfl_xor(mx, 4, 32));
            mx = fmaxf(mx, __shfl_xor(mx, 8, 32));
            float mnew = fmaxf(mrow[r], mx);
            float pe   = __expf(sv[r] - mnew);
            float sum  = pe;
            sum += __shfl_xor(sum, 1, 32);
            sum += __shfl_xor(sum, 2, 32);
            sum += __shfl_xor(sum, 4, 32);
            sum += __shfl_xor(sum, 8, 32);
            float scale = __expf(mrow[r] - mnew);
            lrow[r] = lrow[r] * scale + sum;
            mrow[r] = mnew;
            #pragma unroll
            for (int j = 0; j < 8; ++j) o[j][r] *= scale;
            Pw[(r + rsel) * 18 + lh] = pe;      // D-layout -> A-layout via LDS
        }
        // wave-local LDS store->load across lanes: make writes visible
        asm volatile("s_wait_dscnt 0" ::: "memory");

        // O += P * V  (K-dim = 16 keys -> 4 WMMA steps, 8 output d-tiles)
        const float* Vc = &Vt[cur][0];
        #pragma unroll
        for (int st = 0; st < 4; ++st) {
            int kp = st * 4 + kh2;
            v2f a = *(const v2f*)(&Pw[lh * 18 + kp]);
            #pragma unroll
            for (int j = 0; j < 8; ++j) {
                v2f bfr = *(const v2f*)(&Vc[(j * 16 + lh) * 20 + kp]);  // B=V: key kp,kp+1, dim col
                o[j] = __builtin_amdgcn_wmma_f32_16x16x4_f32(false, a, false, bfr,
                                                             (short)0, o[j], false, false);
            }
        }

        if (kblk < kb_hi) storeKV(cur ^ 1);    // stage next block into other buffer
    }

    // normalize and store ctx in [B, S, HIDDEN]
    #pragma unroll
    for (int r = 0; r < 8; ++r) {
        float inv  = 1.0f / lrow[r];
        int   srow = q0 + w * 16 + r + rsel;
        size_t rowbase = ((size_t)b * SEQ + srow) * HIDDEN + (size_t)h * HEADSZ;
        #pragma unroll
        for (int j = 0; j < 8; ++j)
            ctx[rowbase + j * 16 + lh] = o[j][r] * inv;
    }
}

// ---------------------------------------------------------------------------
extern "C" void kernel_launch(void* const* d_in, const int* in_sizes, int n_in,
                              void* d_out, int out_size, void* d_ws, size_t ws_size,
                              hipStream_t stream)
{
    const float* hs    = (const float*)d_in[0];   // hidden_states [2,2048,2048]
    const float* amask = (const float*)d_in[1];   // attention_mask [2,1,1,2048]
    const float* Wqkv  = (const float*)d_in[2];   // [2048,6144]
    const float* bqkv  = (const float*)d_in[3];   // [6144]
    const float* Wd    = (const float*)d_in[4];   // [2048,2048]
    const float* bd    = (const float*)d_in[5];   // [2048]
    const int*   pos   = (const int*)d_in[6];     // [1,2048]
    float* out = (float*)d_out;

    float* qkv = (float*)d_ws;                                  // 4096 x 6144
    float* ctx = qkv + (size_t)BATCH * SEQ * QKVDIM;            // 4096 x 2048

    (void)in_sizes; (void)n_in; (void)out_size; (void)ws_size;

    // 1) QKV projection: [4096,2048] x [2048,6144] + bias
    gemm_bias_wmma<<<dim3(QKVDIM / 128, (BATCH * SEQ) / 128), dim3(256), 0, stream>>>(
        hs, Wqkv, bqkv, qkv, BATCH * SEQ, QKVDIM, HIDDEN);

    // 2) RoPE in place on q,k rotary dims
    rope_kernel<<<dim3((BATCH * SEQ * NHEADS * 16) / 256), dim3(256), 0, stream>>>(qkv, pos);

    // 3) causal flash attention -> ctx [B,S,HIDDEN]
    flash_attn_kernel<<<dim3(SEQ / 64, BATCH * NHEADS), dim3(128), 0, stream>>>(qkv, amask, ctx);

    // 4) dense projection: [4096,2048] x [2048,2048] + bias -> out
    gemm_bias_wmma<<<dim3(HIDDEN / 128, (BATCH * SEQ) / 128), dim3(256), 0, stream>>>(
        ctx, Wd, bd, out, BATCH * SEQ, HIDDEN, HIDDEN);
}